// TemBlock_1236950581726
// MI455X (gfx1250) — compile-verified
//
#include <hip/hip_runtime.h>
#include <hip/hip_bf16.h>

typedef float v2f __attribute__((ext_vector_type(2)));
typedef float v8f __attribute__((ext_vector_type(8)));

#define NB   16
#define CIN  128
#define CC   256
#define HWN  4096          // 64*64
#define BIG  16777216      // NB*CC*HWN == 4096*4096
#define EPSF 1e-8f

#define WMMA_F32(A, Bv, Cacc) \
    __builtin_amdgcn_wmma_f32_16x16x4_f32(false, (A), false, (Bv), (short)0, (Cacc), false, false)

// ---------------------------------------------------------------------------
// conv1 as 9 shifted 1x1 GEMMs on the matrix pipe.
// Per tap t: C[o][p] += sum_c w1[o,c,t] * x[b,c,shift(p,t)]  (masked at borders)
// A (M=o rows)  : w1 rows (stride 9 in k) -> L2-resident (1.2 MB)
// B (N=p cols)  : x, per-lane spatial position, coalesced across lanes.
//                 Border masking is BRANCHLESS: address clamped to a valid
//                 location, loaded unconditionally, zeroed by a 0/1 float
//                 multiply (no EXEC save/restore in the hot loop).
// Register tile : 4 M-tiles (64 output channels) reuse each B fragment.
// grid (HWN/16, CC/64, NB), 1 wave/block.
// ---------------------------------------------------------------------------
__global__ void __launch_bounds__(32) k_conv1(const float* __restrict__ x,
                                              const float* __restrict__ w1,
                                              const float* __restrict__ b1,
                                              float* __restrict__ out) {
    const int p0 = blockIdx.x * 16;
    const int o0 = blockIdx.y * 64;
    const int b  = blockIdx.z;
    const int lane = threadIdx.x;
    const int half = lane >> 4;
    const int l16  = lane & 15;

    const int p = p0 + l16;          // this lane's spatial column (B operand)
    const int h = p >> 6;
    const int w = p & 63;
    const float* xb = x + (size_t)b * CIN * HWN;

    v8f acc[4] = {};
    for (int t = 0; t < 9; ++t) {
        const int dh = t / 3 - 1;
        const int dw = t % 3 - 1;
        const int hh = h + dh;
        const int ww = w + dw;
        const bool valid = ((unsigned)hh < 64u) && ((unsigned)ww < 64u);
        const float msk = valid ? 1.0f : 0.0f;            // branchless zeroing
        const float* xp = xb + (valid ? (hh * 64 + ww) : 0); // clamped, always safe

        #pragma unroll 4
        for (int ks = 0; ks < CIN; ks += 4) {
            const int ka = ks + half * 2;
            v2f bfrag;
            bfrag.x = xp[(size_t)(ka + 0) * HWN] * msk;   // unconditional load
            bfrag.y = xp[(size_t)(ka + 1) * HWN] * msk;
            #pragma unroll
            for (int mt = 0; mt < 4; ++mt) {
                const float* wr = w1 + (size_t)(o0 + mt * 16 + l16) * (CIN * 9) + t;
                v2f afrag;
                afrag.x = wr[(ka + 0) * 9];
                afrag.y = wr[(ka + 1) * 9];
                acc[mt] = WMMA_F32(afrag, bfrag, acc[mt]);
            }
        }
    }
    float* ob = out + (size_t)b * CC * HWN;
    #pragma unroll
    for (int mt = 0; mt < 4; ++mt) {
        #pragma unroll
        for (int v = 0; v < 8; ++v) {
            const int o = o0 + mt * 16 + v + half * 8;
            ob[(size_t)o * HWN + p0 + l16] = acc[mt][v] + b1[o];
        }
    }
}

// ---------------------------------------------------------------------------
// Mean over HW: one 256-thread block per (b,c) row of 4096.
// ---------------------------------------------------------------------------
__global__ void k_gap(const float* __restrict__ src, float* __restrict__ dst) {
    int bc = blockIdx.x;                             // 0..4095
    const float* p = src + (size_t)bc * HWN;
    float s = 0.f;
    for (int i = threadIdx.x; i < HWN; i += 256) s += p[i];
    __shared__ float red[256];
    red[threadIdx.x] = s;
    __syncthreads();
    for (int w = 128; w >= 1; w >>= 1) {
        if (threadIdx.x < (unsigned)w) red[threadIdx.x] += red[threadIdx.x + w];
        __syncthreads();
    }
    if (threadIdx.x == 0) dst[bc] = red[0] * (1.0f / (float)HWN);
}

// ---------------------------------------------------------------------------
// Tiny dense: out[b][o] = sum_c in[b][c]*w[o][c] + bias[o]. 16x256 outputs.
// ---------------------------------------------------------------------------
__global__ void k_fc(const float* __restrict__ in, const float* __restrict__ w,
                     const float* __restrict__ bias, float* __restrict__ out) {
    int b = blockIdx.x;
    int o = threadIdx.x;
    const float* iv = in + b * CC;
    const float* wr = w + (size_t)o * CC;
    float s = bias[o];
    for (int c = 0; c < CC; ++c) s = fmaf(iv[c], wr[c], s);
    out[b * CC + o] = s;
}

// ---------------------------------------------------------------------------
// add1 = conv1 * (1 + c3), in place.
// ---------------------------------------------------------------------------
__global__ void k_add1(float* __restrict__ a, const float* __restrict__ c3) {
    int t = blockIdx.x * 256 + threadIdx.x;          // 16,777,216
    int bc = t >> 12;                                // b*256 + c
    a[t] *= (1.0f + c3[bc]);
}

// ---------------------------------------------------------------------------
// Cosine similarity along channels: r1[b*4096+p].
// ---------------------------------------------------------------------------
__global__ void k_cos(const float* __restrict__ add1, const float* __restrict__ gap2,
                      float* __restrict__ r1) {
    int t = blockIdx.x * 256 + threadIdx.x;          // 65536
    int b = t >> 12;
    int p = t & (HWN - 1);
    const float* g = gap2 + b * CC;
    const float* A = add1 + (size_t)b * CC * HWN + p;
    float num = 0.f, na = 0.f, ng = 0.f;
    for (int c = 0; c < CC; ++c) {
        float av = A[(size_t)c * HWN];
        float gv = g[c];
        num = fmaf(gv, av, num);
        na  = fmaf(av, av, na);
        ng  = fmaf(gv, gv, ng);
    }
    ng = fmaxf(sqrtf(ng), EPSF);
    na = fmaxf(sqrtf(na), EPSF);
    r1[t] = num / (ng * na);
}

// ---------------------------------------------------------------------------
// m1 = r1^T @ r1 : [4096,4096], K=16. One wave per 16x16 tile (store-bound).
// ---------------------------------------------------------------------------
__global__ void __launch_bounds__(32) k_m1(const float* __restrict__ r1,
                                           float* __restrict__ m1) {
    const int i0 = blockIdx.y * 16;
    const int j0 = blockIdx.x * 16;
    const int lane = threadIdx.x;
    const int half = lane >> 4;
    const int l16  = lane & 15;
    v8f acc = {};
    #pragma unroll
    for (int ks = 0; ks < 16; ks += 4) {
        const int ka = ks + half * 2;
        v2f a, bv;
        a.x  = r1[(size_t)(ka + 0) * HWN + i0 + l16];
        a.y  = r1[(size_t)(ka + 1) * HWN + i0 + l16];
        bv.x = r1[(size_t)(ka + 0) * HWN + j0 + l16];
        bv.y = r1[(size_t)(ka + 1) * HWN + j0 + l16];
        acc = WMMA_F32(a, bv, acc);
    }
    const int col = j0 + l16;
    #pragma unroll
    for (int v = 0; v < 8; ++v) {
        int row = i0 + v + half * 8;
        m1[(size_t)row * HWN + col] = acc[v];
    }
}

// ---------------------------------------------------------------------------
// conv4 (1x1) -> NHWC so the torch permute+view is a free reinterpret.
// nhwc[b][p][o] = sum_c add1[b][c][p] * w4[o][c] + b4[o].
// Register tile: reuse the strided add1 fragment across 4 o-tiles (N=64).
// grid (HWN/16, CC/64, NB), 1 wave/block.
// ---------------------------------------------------------------------------
__global__ void __launch_bounds__(32) k_conv4(const float* __restrict__ add1,
                                              const float* __restrict__ w4,
                                              const float* __restrict__ b4,
                                              float* __restrict__ nhwc) {
    const int p0 = blockIdx.x * 16;
    const int o0 = blockIdx.y * 64;
    const int b  = blockIdx.z;
    const int lane = threadIdx.x;
    const int half = lane >> 4;
    const int l16  = lane & 15;
    const float* Ab = add1 + (size_t)b * CC * HWN + p0 + l16;   // A[m][k]=add1[b,k,p0+m]
    v8f acc[4] = {};
    #pragma unroll 4
    for (int ks = 0; ks < CC; ks += 4) {
        const int ka = ks + half * 2;
        v2f a;
        a.x = Ab[(size_t)(ka + 0) * HWN];
        a.y = Ab[(size_t)(ka + 1) * HWN];
        #pragma unroll
        for (int nt = 0; nt < 4; ++nt) {
            const float* wr = w4 + (size_t)(o0 + nt * 16 + l16) * CC;
            v2f bv = *(const v2f*)(wr + ka);                    // contiguous in K
            acc[nt] = WMMA_F32(a, bv, acc[nt]);
        }
    }
    float* outb = nhwc + (size_t)b * HWN * CC;
    #pragma unroll
    for (int nt = 0; nt < 4; ++nt) {
        const float bias = b4[o0 + nt * 16 + l16];
        #pragma unroll
        for (int v = 0; v < 8; ++v) {
            const int p = p0 + v + half * 8;
            outb[(size_t)p * CC + o0 + nt * 16 + l16] = acc[nt][v] + bias;
        }
    }
}

// ---------------------------------------------------------------------------
// m2[b] = (nhwc[b] as [256 x 4096]) @ m1[4096 x 4096], out in [B,C,H,W].
// Dominant kernel. 4x4 register tile: each wave owns a 64x64 output tile,
// per K-step loads 4 A-frags (float2, contiguous) + 4 B-frags and issues
// 16 WMMAs -> 16 FLOP/B from L2 instead of 4.
// grid (HWN/64, CC/64, NB), 1 wave/block, 16 f32 accumulator tiles.
// ---------------------------------------------------------------------------
__global__ void __launch_bounds__(32) k_m2(const float* __restrict__ nhwc,
                                           const float* __restrict__ m1,
                                           float* __restrict__ out) {
    const int n0 = blockIdx.x * 64;   // 64 tiles over HW
    const int i0 = blockIdx.y * 64;   // 4 tiles over channels
    const int b  = blockIdx.z;
    const int lane = threadIdx.x;
    const int half = lane >> 4;
    const int l16  = lane & 15;

    const float* Abase = nhwc + (size_t)b * CC * HWN;
    const float* Arow[4];
    const float* Bp[4];
    #pragma unroll
    for (int mt = 0; mt < 4; ++mt) Arow[mt] = Abase + (size_t)(i0 + mt * 16 + l16) * HWN;
    #pragma unroll
    for (int nt = 0; nt < 4; ++nt) Bp[nt] = m1 + n0 + nt * 16 + l16;

    __builtin_prefetch(Arow[0], 0, 0);               // global_prefetch_b8
    __builtin_prefetch(Bp[0], 0, 0);

    v8f acc[16] = {};
    #pragma unroll 2
    for (int ks = 0; ks < HWN; ks += 4) {
        const int ka = ks + half * 2;
        v2f a[4], bv[4];
        #pragma unroll
        for (int mt = 0; mt < 4; ++mt) a[mt] = *(const v2f*)(Arow[mt] + ka);
        #pragma unroll
        for (int nt = 0; nt < 4; ++nt) {
            bv[nt].x = Bp[nt][(size_t)(ka + 0) * HWN];
            bv[nt].y = Bp[nt][(size_t)(ka + 1) * HWN];
        }
        #pragma unroll
        for (int mt = 0; mt < 4; ++mt)
            #pragma unroll
            for (int nt = 0; nt < 4; ++nt)
                acc[mt * 4 + nt] = WMMA_F32(a[mt], bv[nt], acc[mt * 4 + nt]);
    }

    float* ob = out + (size_t)b * CC * HWN;
    #pragma unroll
    for (int mt = 0; mt < 4; ++mt) {
        #pragma unroll
        for (int nt = 0; nt < 4; ++nt) {
            #pragma unroll
            for (int v = 0; v < 8; ++v) {
                const int row = i0 + mt * 16 + v + half * 8;
                ob[(size_t)row * HWN + n0 + nt * 16 + l16] = acc[mt * 4 + nt][v];
            }
        }
    }
}

// ---------------------------------------------------------------------------
extern "C" void kernel_launch(void* const* d_in, const int* in_sizes, int n_in,
                              void* d_out, int out_size, void* d_ws, size_t ws_size,
                              hipStream_t stream) {
    const float* x  = (const float*)d_in[0];
    const float* w1 = (const float*)d_in[1];
    const float* b1 = (const float*)d_in[2];
    const float* w2 = (const float*)d_in[3];
    const float* b2 = (const float*)d_in[4];
    const float* w3 = (const float*)d_in[5];
    const float* b3 = (const float*)d_in[6];
    const float* w4 = (const float*)d_in[7];
    const float* b4 = (const float*)d_in[8];
    float* out = (float*)d_out;

    float* ws    = (float*)d_ws;
    float* conv1 = ws;                       // BIG floats (becomes add1 in place)
    float* m1    = conv1 + BIG;              // BIG floats
    float* nhwc  = m1 + BIG;                 // BIG floats
    float* gap1  = nhwc + BIG;               // 4096
    float* c2    = gap1 + NB * CC;           // 4096
    float* c3    = c2 + NB * CC;             // 4096
    float* gap2  = c3 + NB * CC;             // 4096
    float* r1    = gap2 + NB * CC;           // 65536

    // 1) conv1 on the matrix pipe (9 shifted 1x1 GEMMs)
    k_conv1<<<dim3(HWN / 16, CC / 64, NB), dim3(32), 0, stream>>>(x, w1, b1, conv1);
    // 2) gap1
    k_gap<<<dim3(NB * CC), dim3(256), 0, stream>>>(conv1, gap1);
    // 3) c2, c3
    k_fc<<<dim3(NB), dim3(CC), 0, stream>>>(gap1, w2, b2, c2);
    k_fc<<<dim3(NB), dim3(CC), 0, stream>>>(c2, w3, b3, c3);
    // 4) add1 in place
    k_add1<<<dim3(BIG / 256), dim3(256), 0, stream>>>(conv1, c3);
    // 5) gap2
    k_gap<<<dim3(NB * CC), dim3(256), 0, stream>>>(conv1, gap2);
    // 6) cosine -> r1
    k_cos<<<dim3(NB * HWN / 256), dim3(256), 0, stream>>>(conv1, gap2, r1);
    // 7) m1 = r1^T r1  (WMMA, store-bound)
    k_m1<<<dim3(HWN / 16, HWN / 16), dim3(32), 0, stream>>>(r1, m1);
    // 8) conv4 -> NHWC (WMMA, A-fragment reuse x4)
    k_conv4<<<dim3(HWN / 16, CC / 64, NB), dim3(32), 0, stream>>>(conv1, w4, b4, nhwc);
    // 9) m2 = r2 @ m1 -> out (WMMA, 4x4 register tile)
    k_m2<<<dim3(HWN / 64, CC / 64, NB), dim3(32), 0, stream>>>(nhwc, m1, out);
}